// PathPredictor_36060545417339
// MI455X (gfx1250) — compile-verified
//
#include <hip/hip_runtime.h>

#define NN 8192
#define HH 256

typedef float v2f __attribute__((ext_vector_type(2)));
typedef float v8f __attribute__((ext_vector_type(8)));

__device__ __forceinline__ float wave_max(float v) {
#pragma unroll
  for (int d = 16; d; d >>= 1) v = fmaxf(v, __shfl_xor(v, d, 32));
  return v;
}
__device__ __forceinline__ float wave_sum(float v) {
#pragma unroll
  for (int d = 16; d; d >>= 1) v += __shfl_xor(v, d, 32);
  return v;
}

// ---------------- concat 5 x [N,32] -> h0 [N,160] ----------------
__global__ __launch_bounds__(256) void k_concat(
    const float* __restrict__ x, const float* __restrict__ sf,
    const float* __restrict__ tf, const float* __restrict__ og,
    const float* __restrict__ op_, float* __restrict__ h0) {
  int id = blockIdx.x * 256 + threadIdx.x;
  if (id >= NN * 160) return;
  int i = id / 160, c = id - i * 160;
  int g = c >> 5;
  const float* s = (g == 0) ? x : (g == 1) ? sf : (g == 2) ? tf : (g == 3) ? og : op_;
  h0[id] = s[i * 32 + (c & 31)];
}

// ---------------- degree histograms ----------------
__global__ __launch_bounds__(256) void k_hist(
    const int* __restrict__ src, const int* __restrict__ dst, int E,
    int* __restrict__ deg_in, int* __restrict__ deg_out) {
  int e = blockIdx.x * 256 + threadIdx.x;
  if (e >= E) return;
  atomicAdd(&deg_in[dst[e]], 1);
  atomicAdd(&deg_out[src[e]], 1);
}

// -------- exclusive scan of 8192 ints -> rowstart[8193]; grid.x==2 --------
__global__ __launch_bounds__(256) void k_scan(
    const int* __restrict__ degA, int* __restrict__ rsA,
    const int* __restrict__ degB, int* __restrict__ rsB) {
  const int* deg = blockIdx.x ? degB : degA;
  int* rs = blockIdx.x ? rsB : rsA;
  __shared__ int part[256];
  int t = threadIdx.x;
  int base = t * 32;
  int loc[32];
  int s = 0;
#pragma unroll
  for (int k = 0; k < 32; ++k) { loc[k] = s; s += deg[base + k]; }
  part[t] = s;
  __syncthreads();
  int off = 0;
  for (int k = 0; k < t; ++k) off += part[k];
#pragma unroll
  for (int k = 0; k < 32; ++k) rs[base + k] = off + loc[k];
  if (t == 255) rs[NN] = off + s;
}

// ---------------- scatter edges into both CSRs ----------------
__global__ __launch_bounds__(256) void k_scatter(
    const int* __restrict__ src, const int* __restrict__ dst, int E,
    const int* __restrict__ rs_in, const int* __restrict__ rs_out,
    int* __restrict__ cur_in, int* __restrict__ cur_out,
    int* __restrict__ colsrc, int* __restrict__ coldst) {
  int e = blockIdx.x * 256 + threadIdx.x;
  if (e >= E) return;
  int s = src[e], d = dst[e];
  colsrc[rs_in[d] + atomicAdd(&cur_in[d], 1)] = s;
  coldst[rs_out[s] + atomicAdd(&cur_out[s], 1)] = d;
}

// ---- per-row insertion sort (determinism + adjacent-dup dedupe) ----
__global__ __launch_bounds__(256) void k_sortrows(
    const int* __restrict__ rsA, int* __restrict__ colA,
    const int* __restrict__ rsB, int* __restrict__ colB) {
  int id = blockIdx.x * 256 + threadIdx.x;
  const int* rs; int* col; int row;
  if (id < NN)            { rs = rsA; col = colA; row = id; }
  else if (id < 2 * NN)   { rs = rsB; col = colB; row = id - NN; }
  else return;
  int s = rs[row], e = rs[row + 1];
  for (int a = s + 1; a < e; ++a) {
    int v = col[a]; int b = a - 1;
    while (b >= s && col[b] > v) { col[b + 1] = col[b]; --b; }
    col[b + 1] = v;
  }
}

// ---------------- find first row with any nonzero target feature ----------------
__global__ __launch_bounds__(256) void k_target(
    const float* __restrict__ tm, int* __restrict__ tgt) {
  int i = blockIdx.x * 256 + threadIdx.x;
  if (i >= NN) return;
  const float* r = tm + (size_t)i * 32;
  bool nz = false;
#pragma unroll
  for (int k = 0; k < 32; ++k) nz |= (r[k] != 0.0f);
  if (nz) atomicMin(tgt, i);
}

// ---------------- CSR mean aggregation: one wave per node ----------------
template <int W>
__global__ __launch_bounds__(128) void k_aggmean(
    const float* __restrict__ H, const int* __restrict__ rs,
    const int* __restrict__ col, float* __restrict__ out) {
  int lane = threadIdx.x & 31;
  int node = blockIdx.x * 4 + (threadIdx.x >> 5);
  int s = rs[node], e = rs[node + 1];
  constexpr int U = W / 32;
  float acc[U];
#pragma unroll
  for (int u = 0; u < U; ++u) acc[u] = 0.0f;
  for (int p = s; p < e; ++p) {
    const float* hr = H + (size_t)col[p] * W;
#pragma unroll
    for (int u = 0; u < U; ++u) acc[u] += hr[lane + 32 * u];
  }
  int d = e - s;
  float inv = 1.0f / (float)(d > 1 ? d : 1);
  float* o = out + (size_t)node * W;
#pragma unroll
  for (int u = 0; u < U; ++u) o[lane + 32 * u] = acc[u] * inv;
}

// ---------------- SAGE GEMM with fp32 WMMA 16x16x4 ----------------
// Y[i,o] = sum_k Xa[i,k]*Wl[o,k] + sum_k Xb[i,k]*Wr[o,k] + bias[o]
//          (+ bias2[o]) (+ (deg_in[i]>0)*vcond[o]);  relu; Y width = 256.
// Wave computes a 16x64 tile.  A frag: lanes 0-15 -> M=lane, VGPRs=K{0,1};
// lanes 16-31 -> K{2,3}.  B frag mirrored (lane=N).  C/D: VGPR r -> M=r / M=8+r.
__global__ __launch_bounds__(128) void k_gemm_sage(
    const float* __restrict__ Xa, const float* __restrict__ Wl,
    const float* __restrict__ Xb, const float* __restrict__ Wr,
    int K, int ldx, int ldw,
    const float* __restrict__ bias, const float* __restrict__ bias2,
    const float* __restrict__ vcond, const int* __restrict__ degin,
    float* __restrict__ Y) {
  int lane = threadIdx.x & 31;
  int wave = threadIdx.x >> 5;
  int i0 = blockIdx.x * 16;
  int n0 = wave * 64;
  int half = lane >> 4;
  int l16 = lane & 15;
  int akoff = half * 2;
  v8f acc0 = {}, acc1 = {}, acc2 = {}, acc3 = {};
#pragma unroll 1
  for (int pass = 0; pass < 2; ++pass) {
    const float* X = pass ? Xb : Xa;
    const float* Wm = pass ? Wr : Wl;
    const float* xrow = X + (size_t)(i0 + l16) * ldx + akoff;
    const float* w0 = Wm + (size_t)(n0 + l16) * ldw + akoff;
    const float* w1 = w0 + (size_t)16 * ldw;
    const float* w2 = w0 + (size_t)32 * ldw;
    const float* w3 = w0 + (size_t)48 * ldw;
#pragma unroll 4
    for (int k0 = 0; k0 < K; k0 += 4) {
      v2f a  = *(const v2f*)(xrow + k0);
      v2f b0 = *(const v2f*)(w0 + k0);
      v2f b1 = *(const v2f*)(w1 + k0);
      v2f b2 = *(const v2f*)(w2 + k0);
      v2f b3 = *(const v2f*)(w3 + k0);
      acc0 = __builtin_amdgcn_wmma_f32_16x16x4_f32(false, a, false, b0, (short)0, acc0, false, false);
      acc1 = __builtin_amdgcn_wmma_f32_16x16x4_f32(false, a, false, b1, (short)0, acc1, false, false);
      acc2 = __builtin_amdgcn_wmma_f32_16x16x4_f32(false, a, false, b2, (short)0, acc2, false, false);
      acc3 = __builtin_amdgcn_wmma_f32_16x16x4_f32(false, a, false, b3, (short)0, acc3, false, false);
    }
  }
#pragma unroll
  for (int t = 0; t < 4; ++t) {
    v8f acc = (t == 0) ? acc0 : (t == 1) ? acc1 : (t == 2) ? acc2 : acc3;
    int colc = n0 + t * 16 + l16;
    float badd = bias[colc];
    if (bias2) badd += bias2[colc];
    float vc = vcond ? vcond[colc] : 0.0f;
#pragma unroll
    for (int r = 0; r < 8; ++r) {
      int row = i0 + half * 8 + r;
      float val = acc[r] + badd;
      if (vcond && degin[row] > 0) val += vc;
      Y[(size_t)row * HH + colc] = fmaxf(val, 0.0f);
    }
  }
}

// ---------------- row LayerNorm over 256, in place ----------------
__global__ __launch_bounds__(128) void k_layernorm(
    float* __restrict__ H, const float* __restrict__ g, const float* __restrict__ b) {
  int lane = threadIdx.x & 31;
  int row = blockIdx.x * 4 + (threadIdx.x >> 5);
  float* h = H + (size_t)row * HH;
  float v[8];
  float s = 0.0f;
#pragma unroll
  for (int u = 0; u < 8; ++u) { v[u] = h[lane + 32 * u]; s += v[u]; }
  s = wave_sum(s);
  float mean = s * (1.0f / 256.0f);
  float var = 0.0f;
#pragma unroll
  for (int u = 0; u < 8; ++u) { float d = v[u] - mean; var += d * d; }
  var = wave_sum(var) * (1.0f / 256.0f);
  float inv = 1.0f / sqrtf(var + 1e-5f);
#pragma unroll
  for (int u = 0; u < 8; ++u)
    h[lane + 32 * u] = (v[u] - mean) * inv * g[lane + 32 * u] + b[lane + 32 * u];
}

// ---- u2[o] = tfeat . wr2[o,256:512] ; v2[o] = tfeat . wl2[o,256:512] ----
__global__ __launch_bounds__(256) void k_u2v2(
    const float* __restrict__ h1, const int* __restrict__ tgt,
    const float* __restrict__ wl2, const float* __restrict__ wr2,
    float* __restrict__ u2, float* __restrict__ v2) {
  __shared__ float tf[HH];
  int t = *tgt;
  if (t < 0 || t >= NN) t = 0;   // reference argmax -> 0 if no nonzero row
  int o = threadIdx.x;
  tf[o] = h1[(size_t)t * HH + o];
  __syncthreads();
  const float* wr = wr2 + (size_t)o * 512 + 256;
  const float* wl = wl2 + (size_t)o * 512 + 256;
  float su = 0.0f, sv = 0.0f;
  for (int k = 0; k < HH; ++k) { su += tf[k] * wr[k]; sv += tf[k] * wl[k]; }
  u2[o] = su;
  v2[o] = sv;
}

// ---- masked softmax: full softmax denom cancels under neighbor mask ----
// One wave per row; neighbors from sorted CSR-by-src (dedupe adjacent + self).
__global__ __launch_bounds__(128) void k_masked_softmax(
    const float* __restrict__ h3, const float* __restrict__ fcw,
    const float* __restrict__ fcb, const int* __restrict__ rs_out,
    const int* __restrict__ coldst, float* __restrict__ out) {
  __shared__ float sh[4][HH];
  int lane = threadIdx.x & 31;
  int wave = threadIdx.x >> 5;
  int row = blockIdx.x * 4 + wave;
  const float* hr = h3 + (size_t)row * HH;
#pragma unroll
  for (int u = 0; u < 8; ++u) sh[wave][lane + 32 * u] = hr[lane + 32 * u];
  __syncthreads();
  const float* hs = sh[wave];
  int s = rs_out[row], e = rs_out[row + 1];
  float lv[8]; int jv[8]; int cnt = 0;
  float mloc = -3.4e38f;
  for (int p = s + lane; p < e; p += 32) {
    int j = coldst[p];
    bool dup = (j == row) || (p > s && coldst[p - 1] == j);
    if (!dup && cnt < 8) {
      const float* w = fcw + (size_t)j * HH;
      float acc = fcb[j];
      for (int k = 0; k < HH; k += 4) {
        float4 wv = *(const float4*)(w + k);
        acc += hs[k] * wv.x + hs[k + 1] * wv.y + hs[k + 2] * wv.z + hs[k + 3] * wv.w;
      }
      lv[cnt] = acc; jv[cnt] = j; ++cnt;
      mloc = fmaxf(mloc, acc);
    }
  }
  float ldiag = 0.0f;
  if (lane == 0) {  // diagonal always in the mask
    const float* w = fcw + (size_t)row * HH;
    float acc = fcb[row];
    for (int k = 0; k < HH; k += 4) {
      float4 wv = *(const float4*)(w + k);
      acc += hs[k] * wv.x + hs[k + 1] * wv.y + hs[k + 2] * wv.z + hs[k + 3] * wv.w;
    }
    ldiag = acc;
    mloc = fmaxf(mloc, acc);
  }
  float m = wave_max(mloc);
  float ssum = 0.0f;
  for (int c = 0; c < cnt; ++c) ssum += expf(lv[c] - m);
  if (lane == 0) ssum += expf(ldiag - m);
  ssum = wave_sum(ssum);
  float inv = 1.0f / ssum;
  float* orow = out + (size_t)row * NN;
  for (int c = 0; c < cnt; ++c) orow[jv[c]] = expf(lv[c] - m) * inv;
  if (lane == 0) orow[row] = expf(ldiag - m) * inv;
}

extern "C" void kernel_launch(void* const* d_in, const int* in_sizes, int n_in,
                              void* d_out, int out_size, void* d_ws, size_t ws_size,
                              hipStream_t stream) {
  const float* x   = (const float*)d_in[0];
  const float* sf  = (const float*)d_in[1];
  const float* tfm = (const float*)d_in[2];
  const float* og  = (const float*)d_in[3];
  const float* op_ = (const float*)d_in[4];
  const float* w1l = (const float*)d_in[5];
  const float* w1r = (const float*)d_in[6];
  const float* b1  = (const float*)d_in[7];
  const float* w2l = (const float*)d_in[8];
  const float* w2r = (const float*)d_in[9];
  const float* b2  = (const float*)d_in[10];
  const float* w3l = (const float*)d_in[11];
  const float* w3r = (const float*)d_in[12];
  const float* b3  = (const float*)d_in[13];
  const float* fcw = (const float*)d_in[14];
  const float* fcb = (const float*)d_in[15];
  const float* g1  = (const float*)d_in[16];
  const float* lb1 = (const float*)d_in[17];
  const float* g2  = (const float*)d_in[18];
  const float* lb2 = (const float*)d_in[19];
  const int*   ei  = (const int*)d_in[20];
  int E = in_sizes[20] / 2;
  const int* esrc = ei;
  const int* edst = ei + E;

  char* ws = (char*)d_ws;
  size_t off = 0;
  auto allocf = [&](size_t n) { float* p = (float*)(ws + off); off += n * 4; return p; };
  auto alloci = [&](size_t n) { int* p = (int*)(ws + off); off += n * 4; return p; };
  float* h0  = allocf((size_t)NN * 160);
  float* h1  = allocf((size_t)NN * HH);
  float* agg = allocf((size_t)NN * HH);
  float* h2  = allocf((size_t)NN * HH);
  float* h3  = allocf((size_t)NN * HH);
  float* u2  = allocf(HH);
  float* v2  = allocf(HH);
  int* deg_in  = alloci(NN);   // deg_in..cur_out contiguous -> single memset
  int* deg_out = alloci(NN);
  int* cur_in  = alloci(NN);
  int* cur_out = alloci(NN);
  int* rs_in   = alloci(NN + 1);
  int* rs_out  = alloci(NN + 1);
  int* colsrc  = alloci(E);
  int* coldst  = alloci(E);
  int* tgt     = alloci(1);
  (void)ws_size; (void)n_in;

  hipMemsetAsync(deg_in, 0, sizeof(int) * 4 * NN, stream);
  hipMemsetAsync(tgt, 0x7f, sizeof(int), stream);
  hipMemsetAsync(d_out, 0, (size_t)out_size * sizeof(float), stream);

  k_concat<<<(NN * 160 + 255) / 256, 256, 0, stream>>>(x, sf, tfm, og, op_, h0);
  k_target<<<NN / 256, 256, 0, stream>>>(tfm, tgt);
  k_hist<<<(E + 255) / 256, 256, 0, stream>>>(esrc, edst, E, deg_in, deg_out);
  k_scan<<<2, 256, 0, stream>>>(deg_in, rs_in, deg_out, rs_out);
  k_scatter<<<(E + 255) / 256, 256, 0, stream>>>(esrc, edst, E, rs_in, rs_out,
                                                 cur_in, cur_out, colsrc, coldst);
  k_sortrows<<<(2 * NN + 255) / 256, 256, 0, stream>>>(rs_in, colsrc, rs_out, coldst);

  // conv1: [N,160] -> [N,256], relu, LN1
  k_aggmean<160><<<NN / 4, 128, 0, stream>>>(h0, rs_in, colsrc, agg);
  k_gemm_sage<<<NN / 16, 128, 0, stream>>>(agg, w1l, h0, w1r, 160, 160, 160,
                                           b1, nullptr, nullptr, nullptr, h1);
  k_layernorm<<<NN / 4, 128, 0, stream>>>(h1, g1, lb1);

  // conv2: concat with broadcast tfeat folded into u2/v2 constants
  k_u2v2<<<1, 256, 0, stream>>>(h1, tgt, w2l, w2r, u2, v2);
  k_aggmean<256><<<NN / 4, 128, 0, stream>>>(h1, rs_in, colsrc, agg);
  k_gemm_sage<<<NN / 16, 128, 0, stream>>>(agg, w2l, h1, w2r, 256, 256, 512,
                                           b2, u2, v2, deg_in, h2);
  k_layernorm<<<NN / 4, 128, 0, stream>>>(h2, g2, lb2);

  // conv3
  k_aggmean<256><<<NN / 4, 128, 0, stream>>>(h2, rs_in, colsrc, agg);
  k_gemm_sage<<<NN / 16, 128, 0, stream>>>(agg, w3l, h2, w3r, 256, 256, 256,
                                           b3, nullptr, nullptr, nullptr, h3);

  // masked softmax over neighbor columns + self (dense softmax cancels)
  k_masked_softmax<<<NN / 4, 128, 0, stream>>>(h3, fcw, fcb, rs_out, coldst,
                                               (float*)d_out);
}